// FocalLoss_69758858822546
// MI455X (gfx1250) — compile-verified
//
#include <hip/hip_runtime.h>
#include <hip/hip_bf16.h>
#include <stdint.h>

// Focal loss, N=8192 rows x C=32000 classes, f32.
// Roofline: 8192*32000*4 B = 1.05 GB read once -> ~45us at 23.3 TB/s. No
// matmul structure => WMMA irrelevant; the MI455X win is the async
// global->LDS pipeline (ASYNCcnt) feeding a single-sweep, numerically
// stable softmax reduction:
//   - one block per row, 256 threads (8 wave32)
//   - 2 x 16KB static LDS buffers, double-buffered via
//     global_load_async_to_lds_b128 + partial s_wait_asynccnt(4)
//     (exactly 4 async issues per wave per chunk; in-order completion
//      makes cnt<=4 mean "previous chunk resident")
//   - per chunk: max over the thread's 16 staged values, rescale running
//     sum once, then 1 exp/element  => ~1 exp per element total, well
//     under the HBM roofline
//   - (m,s) merged with log-sum-exp combine: wave32 shfl_xor, then LDS
// Deterministic 2-stage mean (no float atomics).

#define BLOCK 256
#define CHUNK_V4 1024                   // vec4 slots per chunk (16 KB)
#define LOADS_PER_CHUNK 4               // CHUNK_V4 / BLOCK

#if defined(__gfx1250__) && \
    __has_builtin(__builtin_amdgcn_global_load_async_to_lds_b128) && \
    __has_builtin(__builtin_amdgcn_s_wait_asynccnt)
#define USE_ASYNC_LDS 1
#else
#define USE_ASYNC_LDS 0
#endif

typedef int v4i __attribute__((vector_size(16)));
typedef __attribute__((address_space(1))) v4i gv4i;   // global src
typedef __attribute__((address_space(3))) v4i lv4i;   // LDS dst

// Issue one chunk's staging copies. Addresses are clamped to the last valid
// vec4 so every wave issues exactly LOADS_PER_CHUNK async instructions
// (uniform ASYNCcnt); clamped lanes redundantly write the same value to the
// same LDS slot (benign).
__device__ __forceinline__ void issue_chunk(const char* gbase, float* lbuf,
                                            int cbase, int nvec, int tid)
{
    #pragma unroll
    for (int k = 0; k < LOADS_PER_CHUNK; ++k) {
        int v  = cbase + tid + (k << 8);
        int vc = (v < nvec) ? v : (nvec - 1);
#if USE_ASYNC_LDS
        __builtin_amdgcn_global_load_async_to_lds_b128(
            (gv4i*)(gbase + (size_t)vc * 16),
            (lv4i*)((char*)lbuf + (size_t)(vc - cbase) * 16),
            /*imm offset*/0, /*cpol*/0);
#else
        *(float4*)((char*)lbuf + (size_t)(vc - cbase) * 16) =
            *(const float4*)(gbase + (size_t)vc * 16);
#endif
    }
}

__device__ __forceinline__ void wait_chunk_prev_resident()
{
#if USE_ASYNC_LDS
    __builtin_amdgcn_s_wait_asynccnt(LOADS_PER_CHUNK);  // older chunk done
#endif
}
__device__ __forceinline__ void wait_chunk_all_resident()
{
#if USE_ASYNC_LDS
    __builtin_amdgcn_s_wait_asynccnt(0);
#endif
}

__global__ void __launch_bounds__(BLOCK)
focal_row_kernel(const float* __restrict__ inp,
                 const long long* __restrict__ target,   // reference: int64
                 const float* __restrict__ weight,
                 float* __restrict__ row_loss,
                 int C)
{
    __shared__ float ldsA[CHUNK_V4 * 4];   // 16 KB
    __shared__ float ldsB[CHUNK_V4 * 4];   // 16 KB
    __shared__ float sredM[BLOCK / 32];
    __shared__ float sredS[BLOCK / 32];

    const int row  = blockIdx.x;
    const int tid  = threadIdx.x;
    const int lane = tid & 31;
    const int wave = tid >> 5;

    const float* g  = inp + (size_t)row * (size_t)C;
    const char*  gb = (const char*)g;
    const int nvec = C >> 2;                             // 8000
    const int nch  = (nvec + CHUNK_V4 - 1) / CHUNK_V4;   // 8

    float m = -INFINITY;   // running max (this thread's elements)
    float s = 0.0f;        // running sum of exp(x - m)

    issue_chunk(gb, ldsA, 0, nvec, tid);                 // prime the pipe

    for (int c = 0; c < nch; ++c) {
        const int cbase = c * CHUNK_V4;
        // Prefetch next chunk into the other buffer, then wait until the
        // *current* chunk's loads (the older LOADS_PER_CHUNK) have landed.
        if (c + 1 < nch) {
            issue_chunk(gb, ((c + 1) & 1) ? ldsB : ldsA,
                        cbase + CHUNK_V4, nvec, tid);
            wait_chunk_prev_resident();
        } else {
            wait_chunk_all_resident();
        }
        __syncthreads();   // all waves' copies for chunk c now in LDS

        // ---- consume chunk c: per-thread max of its 16 values, one
        //      rescale of the running sum, then 1 exp per element ----
        const float4* buf = (const float4*)((c & 1) ? ldsB : ldsA);
        const int climit = (cbase + CHUNK_V4 < nvec) ? (cbase + CHUNK_V4) : nvec;

        float4 vals[LOADS_PER_CHUNK];
        bool   ok[LOADS_PER_CHUNK];
        float  lm = -INFINITY;
        #pragma unroll
        for (int k = 0; k < LOADS_PER_CHUNK; ++k) {
            int v = cbase + tid + (k << 8);
            ok[k] = (v < climit);
            if (ok[k]) {
                float4 x = buf[v - cbase];
                vals[k] = x;
                lm = fmaxf(lm, fmaxf(fmaxf(x.x, x.y), fmaxf(x.z, x.w)));
            }
        }
        const float mn = fmaxf(m, lm);
        s *= __expf(m - mn);          // m=-INF first chunk -> exp(-inf)=0, s stays 0
        m = mn;
        #pragma unroll
        for (int k = 0; k < LOADS_PER_CHUNK; ++k) {
            if (ok[k]) {
                float4 x = vals[k];
                s += __expf(x.x - m) + __expf(x.y - m) +
                     __expf(x.z - m) + __expf(x.w - m);
            }
        }
        __syncthreads();   // done reading buf before it is re-targeted
    }

    // ---- merge (m, s) across the wave (log-sum-exp combine) ----
    #pragma unroll
    for (int o = 16; o > 0; o >>= 1) {
        float mo = __shfl_xor(m, o, 32);
        float so = __shfl_xor(s, o, 32);
        float M  = fmaxf(m, mo);
        s = s * __expf(m - M) + so * __expf(mo - M);
        m = M;
    }
    if (lane == 0) { sredM[wave] = m; sredS[wave] = s; }
    __syncthreads();

    if (tid == 0) {
        float M = sredM[0], S = sredS[0];
        #pragma unroll
        for (int w = 1; w < BLOCK / 32; ++w) {
            float mo = sredM[w], so = sredS[w];
            float Mn = fmaxf(M, mo);
            S = S * __expf(M - Mn) + so * __expf(mo - Mn);
            M = Mn;
        }
        const long long t = target[row];
        const float xt    = g[(int)t];                 // single L2-hit load
        const float logpt = (xt - M) - __logf(S);      // log_softmax at target
        const float pt    = __expf(logpt);
        const float omp   = 1.0f - pt;
        row_loss[row] = -weight[(int)t] * omp * omp * logpt;   // GAMMA = 2
    }
}

__device__ __forceinline__ float wave_sum(float v) {
    #pragma unroll
    for (int o = 16; o > 0; o >>= 1) v += __shfl_xor(v, o, 32);
    return v;
}

__global__ void __launch_bounds__(BLOCK)
mean_reduce_kernel(const float* __restrict__ v, float* __restrict__ out, int n)
{
    __shared__ float sred[BLOCK / 32];
    const int tid = threadIdx.x, lane = tid & 31, wave = tid >> 5;
    float s = 0.0f;
    for (int i = tid; i < n; i += BLOCK) s += v[i];
    s = wave_sum(s);
    if (lane == 0) sred[wave] = s;
    __syncthreads();
    if (tid == 0) {
        float tot = 0.0f;
        #pragma unroll
        for (int w = 0; w < BLOCK / 32; ++w) tot += sred[w];
        out[0] = tot / (float)n;
    }
}

extern "C" void kernel_launch(void* const* d_in, const int* in_sizes, int n_in,
                              void* d_out, int out_size, void* d_ws, size_t ws_size,
                              hipStream_t stream)
{
    const float*     inp    = (const float*)d_in[0];       // [N, C] f32
    const long long* target = (const long long*)d_in[1];   // [N] int64
    const float*     weight = (const float*)d_in[2];       // [C] f32
    const int N = in_sizes[1];                             // 8192
    const int C = in_sizes[2];                             // 32000

    float* row_loss = (float*)d_ws;                        // N floats scratch

    focal_row_kernel<<<N, BLOCK, 0, stream>>>(inp, target, weight, row_loss, C);
    mean_reduce_kernel<<<1, BLOCK, 0, stream>>>(row_loss, (float*)d_out, N);
}